// GAT_9706626090119
// MI455X (gfx1250) — compile-verified
//
#include <hip/hip_runtime.h>
#include <hip/hip_bf16.h>

// GATv2 GNN forward for MI455X (gfx1250), wave32, f32 WMMA 16x16x4 path.
//
// Input layout assumption (setup_inputs dict order, params flattened in
// insertion order):
//  0 x(N*5)  1 edge_index(2E,int32)  2 edge_attr(E)
//  3..6   comm.l1.W(128x64) comm.l1.b comm.l2.W(64x64) comm.l2.b
//  7..10  toll.l1.W(64x64)  toll.l1.b toll.l2.W(64x64) toll.l2.b
// 11..14  edge.l1.W(64x64)  edge.l1.b edge.l2.W(64x64) edge.l2.b
// 15..21  conv1: Wl(64x192) bl Wr(64x192) br We(64x192) att(3x64) bias(192)
// 22..28  conv2: Wl(192x192) bl Wr(192x192) br We(64x192) att(3x64) bias(192)
// 29..35  conv3: Wl(192x64) bl Wr(192x64) br We(64x64) att(64) bias(64)
// 36..39  out.l1.W(64x32) out.l1.b(32) out.l2.W(32) out.l2.b(1)

typedef __attribute__((ext_vector_type(2))) float v2f;
typedef __attribute__((ext_vector_type(8))) float v8f;

__device__ __forceinline__ v8f wmma_f32(v2f a, v2f b, v8f c) {
  // D = A(16x4,f32) * B(4x16,f32) + C(16x16,f32), exact f32 matrix core op.
  return __builtin_amdgcn_wmma_f32_16x16x4_f32(false, a, false, b,
                                               (short)0, c, false, false);
}

__device__ __forceinline__ float fe_of(int j) {
  float t = (float)j * (1.0f / 63.0f);
  return 10.0f * t * t; // SCALE * linspace(0,1,64)^2
}

__device__ __forceinline__ void atomicMaxF(float* addr, float val) {
  // monotonic float-bits trick; buffer initialized to -inf
  if (val >= 0.0f)
    atomicMax((int*)addr, __float_as_int(val));
  else
    atomicMin((unsigned int*)addr, (unsigned int)__float_as_uint(val));
}

// ---------------------------------------------------------------- utilities
__global__ void fill_kernel(float* p, size_t n, float v) {
  size_t i = (size_t)blockIdx.x * blockDim.x + threadIdx.x;
  if (i < n) p[i] = v;
}

// ---------------------------------------------- node feature prep + 2 MLPs
// One wave (block of 32) per 16-node tile. WMMA f32 16x16x4 throughout.
__global__ void node_mlp_kernel(const float* __restrict__ x,
                                const float* __restrict__ W1c, const float* __restrict__ b1c,
                                const float* __restrict__ W2c, const float* __restrict__ b2c,
                                const float* __restrict__ W1t, const float* __restrict__ b1t,
                                const float* __restrict__ W2t, const float* __restrict__ b2t,
                                float* __restrict__ h0, int N) {
  __shared__ float stage[16 * 65]; // padded to kill LDS bank conflicts
  int lane = threadIdx.x & 31;
  int laneM = lane & 15, half = lane >> 4;
  int tile = blockIdx.x;
  int nd = tile * 16 + laneM;
  if (nd >= N) nd = N - 1;
  float x1 = x[(size_t)nd * 5 + 1];
  float x2 = x[(size_t)nd * 5 + 2];
  float x3 = x[(size_t)nd * 5 + 3];

  // ---- comm path: layer1 K=128 (xc = [x1*fe, x2*fe]), dout=64
  v8f acc[4];
  #pragma unroll
  for (int nt = 0; nt < 4; nt++) { v8f z = {}; acc[nt] = z; }
  for (int k0 = 0; k0 < 128; k0 += 4) {
    int ka = k0 + 2 * half;
    v2f a;
    a.x = (ka < 64) ? x1 * fe_of(ka) : x2 * fe_of(ka - 64);
    a.y = (ka + 1 < 64) ? x1 * fe_of(ka + 1) : x2 * fe_of(ka + 1 - 64);
    #pragma unroll
    for (int nt = 0; nt < 4; nt++) {
      v2f b;
      b.x = W1c[ka * 64 + nt * 16 + laneM];
      b.y = W1c[(ka + 1) * 64 + nt * 16 + laneM];
      acc[nt] = wmma_f32(a, b, acc[nt]);
    }
  }
  #pragma unroll
  for (int nt = 0; nt < 4; nt++)
    #pragma unroll
    for (int r = 0; r < 8; r++)
      stage[(r + 8 * half) * 65 + nt * 16 + laneM] =
          fmaxf(acc[nt][r] + b1c[nt * 16 + laneM], 0.0f);
  __syncthreads();

  // comm layer2: K=64
  v8f accC[4];
  #pragma unroll
  for (int nt = 0; nt < 4; nt++) { v8f z = {}; accC[nt] = z; }
  for (int k0 = 0; k0 < 64; k0 += 4) {
    int ka = k0 + 2 * half;
    v2f a;
    a.x = stage[laneM * 65 + ka];
    a.y = stage[laneM * 65 + ka + 1];
    #pragma unroll
    for (int nt = 0; nt < 4; nt++) {
      v2f b;
      b.x = W2c[ka * 64 + nt * 16 + laneM];
      b.y = W2c[(ka + 1) * 64 + nt * 16 + laneM];
      accC[nt] = wmma_f32(a, b, accC[nt]);
    }
  }
  __syncthreads();

  // ---- toll path: layer1 K=64 (xt = x3*fe)
  #pragma unroll
  for (int nt = 0; nt < 4; nt++) { v8f z = {}; acc[nt] = z; }
  for (int k0 = 0; k0 < 64; k0 += 4) {
    int ka = k0 + 2 * half;
    v2f a;
    a.x = x3 * fe_of(ka);
    a.y = x3 * fe_of(ka + 1);
    #pragma unroll
    for (int nt = 0; nt < 4; nt++) {
      v2f b;
      b.x = W1t[ka * 64 + nt * 16 + laneM];
      b.y = W1t[(ka + 1) * 64 + nt * 16 + laneM];
      acc[nt] = wmma_f32(a, b, acc[nt]);
    }
  }
  #pragma unroll
  for (int nt = 0; nt < 4; nt++)
    #pragma unroll
    for (int r = 0; r < 8; r++)
      stage[(r + 8 * half) * 65 + nt * 16 + laneM] =
          fmaxf(acc[nt][r] + b1t[nt * 16 + laneM], 0.0f);
  __syncthreads();

  v8f accT[4];
  #pragma unroll
  for (int nt = 0; nt < 4; nt++) { v8f z = {}; accT[nt] = z; }
  for (int k0 = 0; k0 < 64; k0 += 4) {
    int ka = k0 + 2 * half;
    v2f a;
    a.x = stage[laneM * 65 + ka];
    a.y = stage[laneM * 65 + ka + 1];
    #pragma unroll
    for (int nt = 0; nt < 4; nt++) {
      v2f b;
      b.x = W2t[ka * 64 + nt * 16 + laneM];
      b.y = W2t[(ka + 1) * 64 + nt * 16 + laneM];
      accT[nt] = wmma_f32(a, b, accT[nt]);
    }
  }

  // combine: h = comm*(1-mask) + toll*mask
  #pragma unroll
  for (int nt = 0; nt < 4; nt++)
    #pragma unroll
    for (int r = 0; r < 8; r++) {
      int orow = tile * 16 + r + 8 * half;
      if (orow < N) {
        float mk = x[(size_t)orow * 5];
        float c = accC[nt][r] + b2c[nt * 16 + laneM];
        float t = accT[nt][r] + b2t[nt * 16 + laneM];
        h0[(size_t)orow * 64 + nt * 16 + laneM] = c * (1.0f - mk) + t * mk;
      }
    }
}

// ----------------------------------------------------------- edge MLP (WMMA)
__global__ void edge_mlp_kernel(const float* __restrict__ edge_attr,
                                const float* __restrict__ W1, const float* __restrict__ b1,
                                const float* __restrict__ W2, const float* __restrict__ b2,
                                float* __restrict__ eemb, int E) {
  __shared__ float stage[16 * 65];
  int lane = threadIdx.x & 31;
  int laneM = lane & 15, half = lane >> 4;
  int tile = blockIdx.x;
  int er = tile * 16 + laneM;
  if (er >= E) er = E - 1;
  float ea = edge_attr[er];

  v8f acc[4];
  #pragma unroll
  for (int nt = 0; nt < 4; nt++) { v8f z = {}; acc[nt] = z; }
  for (int k0 = 0; k0 < 64; k0 += 4) {
    int ka = k0 + 2 * half;
    v2f a;
    a.x = ea * fe_of(ka);
    a.y = ea * fe_of(ka + 1);
    #pragma unroll
    for (int nt = 0; nt < 4; nt++) {
      v2f b;
      b.x = W1[ka * 64 + nt * 16 + laneM];
      b.y = W1[(ka + 1) * 64 + nt * 16 + laneM];
      acc[nt] = wmma_f32(a, b, acc[nt]);
    }
  }
  #pragma unroll
  for (int nt = 0; nt < 4; nt++)
    #pragma unroll
    for (int r = 0; r < 8; r++)
      stage[(r + 8 * half) * 65 + nt * 16 + laneM] =
          fmaxf(acc[nt][r] + b1[nt * 16 + laneM], 0.0f);
  __syncthreads();

  v8f acc2[4];
  #pragma unroll
  for (int nt = 0; nt < 4; nt++) { v8f z = {}; acc2[nt] = z; }
  for (int k0 = 0; k0 < 64; k0 += 4) {
    int ka = k0 + 2 * half;
    v2f a;
    a.x = stage[laneM * 65 + ka];
    a.y = stage[laneM * 65 + ka + 1];
    #pragma unroll
    for (int nt = 0; nt < 4; nt++) {
      v2f b;
      b.x = W2[ka * 64 + nt * 16 + laneM];
      b.y = W2[(ka + 1) * 64 + nt * 16 + laneM];
      acc2[nt] = wmma_f32(a, b, acc2[nt]);
    }
  }
  #pragma unroll
  for (int nt = 0; nt < 4; nt++)
    #pragma unroll
    for (int r = 0; r < 8; r++) {
      int orow = tile * 16 + r + 8 * half;
      if (orow < E)
        eemb[(size_t)orow * 64 + nt * 16 + laneM] =
            acc2[nt][r] + b2[nt * 16 + laneM];
    }
}

// -------------------------------------------------- self-loop attr (scatter)
__global__ void loop_scatter_kernel(const int* __restrict__ tgt,
                                    const float* __restrict__ eemb,
                                    float* __restrict__ loopat,
                                    float* __restrict__ cnt, int E) {
  size_t idx = (size_t)blockIdx.x * blockDim.x + threadIdx.x;
  size_t total = (size_t)E * 64;
  if (idx >= total) return;
  int e = (int)(idx >> 6);
  int c = (int)(idx & 63);
  int t = tgt[e];
  atomicAdd(&loopat[(size_t)t * 64 + c], eemb[idx]);
  if (c == 0) atomicAdd(&cnt[t], 1.0f);
}

__global__ void loop_div_kernel(float* __restrict__ loopat,
                                const float* __restrict__ cnt, int N) {
  size_t idx = (size_t)blockIdx.x * blockDim.x + threadIdx.x;
  if (idx >= (size_t)N * 64) return;
  loopat[idx] /= fmaxf(cnt[idx >> 6], 1.0f);
}

// ------------------------------------------- generic GEMM+bias (WMMA, f32)
// out(N x DOUT) = A(N x DIN) @ W(DIN x DOUT) + bias
template <int DIN, int DOUT>
__global__ void gemm_bias_kernel(const float* __restrict__ A,
                                 const float* __restrict__ W,
                                 const float* __restrict__ bias,
                                 float* __restrict__ out, int nrows) {
  constexpr int NT = DOUT / 16;
  int ntiles = (nrows + 15) / 16;
  int wave = blockIdx.x * (blockDim.x >> 5) + (threadIdx.x >> 5);
  if (wave >= ntiles) return;
  int lane = threadIdx.x & 31;
  int laneM = lane & 15, half = lane >> 4;
  int row = wave * 16 + laneM;
  if (row >= nrows) row = nrows - 1;
  const float* arow = A + (size_t)row * DIN;

  v8f acc[NT];
  #pragma unroll
  for (int nt = 0; nt < NT; nt++) { v8f z = {}; acc[nt] = z; }

  for (int k0 = 0; k0 < DIN; k0 += 4) {
    int ka = k0 + 2 * half;
    v2f a;
    a.x = arow[ka];
    a.y = arow[ka + 1];
    #pragma unroll
    for (int nt = 0; nt < NT; nt++) {
      v2f b;
      b.x = W[ka * DOUT + nt * 16 + laneM];
      b.y = W[(ka + 1) * DOUT + nt * 16 + laneM];
      acc[nt] = wmma_f32(a, b, acc[nt]);
    }
  }
  #pragma unroll
  for (int nt = 0; nt < NT; nt++)
    #pragma unroll
    for (int r = 0; r < 8; r++) {
      int orow = wave * 16 + r + 8 * half;
      if (orow < nrows)
        out[(size_t)orow * DOUT + nt * 16 + laneM] =
            acc[nt][r] + bias[nt * 16 + laneM];
    }
}

// -------------------------------- GAT pass 1: logits + segment max (atomic)
template <int HEADS>
__global__ void gat_logit_kernel(const int* __restrict__ src, const int* __restrict__ tgt,
                                 int E, int N,
                                 const float* __restrict__ eemb,
                                 const float* __restrict__ loopat,
                                 const float* __restrict__ xl, const float* __restrict__ xr,
                                 const float* __restrict__ We, const float* __restrict__ att,
                                 float* __restrict__ logit, float* __restrict__ m) {
  constexpr int HC = HEADS * 64;
  constexpr int TPL = HC / 32; // cols per lane
  __shared__ float sWe[64 * HC];
  __shared__ float sAtt[HC];
  for (int i = threadIdx.x; i < 64 * HC; i += blockDim.x) sWe[i] = We[i];
  for (int i = threadIdx.x; i < HC; i += blockDim.x) sAtt[i] = att[i];
  __syncthreads();

  int lane = threadIdx.x & 31;
  int wave = blockIdx.x * (blockDim.x >> 5) + (threadIdx.x >> 5);
  int nw = gridDim.x * (blockDim.x >> 5);
  int Ef = E + N;
  for (int ef = wave; ef < Ef; ef += nw) {
    int sN, tN;
    const float* erow;
    if (ef < E) {
      sN = src[ef]; tN = tgt[ef];
      erow = eemb + (size_t)ef * 64;
    } else {
      sN = tN = ef - E;
      erow = loopat + (size_t)sN * 64;
    }
    // ee = e_row(64) @ We(64 x HC), this lane's columns lane+32*tt
    float ee[TPL];
    #pragma unroll
    for (int tt = 0; tt < TPL; tt++) ee[tt] = 0.0f;
    #pragma unroll 8
    for (int k = 0; k < 64; k++) {
      float ev = erow[k];
      #pragma unroll
      for (int tt = 0; tt < TPL; tt++)
        ee[tt] += ev * sWe[k * HC + lane + 32 * tt];
    }
    float p[HEADS];
    #pragma unroll
    for (int h = 0; h < HEADS; h++) p[h] = 0.0f;
    #pragma unroll
    for (int tt = 0; tt < TPL; tt++) {
      int j = lane + 32 * tt; // head = tt/2 (compile-time)
      float v = xl[(size_t)sN * HC + j] + xr[(size_t)tN * HC + j] + ee[tt];
      v = v > 0.0f ? v : 0.2f * v; // leaky_relu(0.2)
      p[tt >> 1] += sAtt[j] * v;
    }
    #pragma unroll
    for (int off = 16; off > 0; off >>= 1)
      #pragma unroll
      for (int h = 0; h < HEADS; h++) p[h] += __shfl_xor(p[h], off, 32);
    if (lane == 0) {
      #pragma unroll
      for (int h = 0; h < HEADS; h++) {
        logit[(size_t)ef * HEADS + h] = p[h];
        atomicMaxF(&m[(size_t)tN * HEADS + h], p[h]);
      }
    }
  }
}

// --------------------------------- GAT pass 2: exp + segment sum (atomic)
template <int HEADS>
__global__ void gat_expsum_kernel(const int* __restrict__ tgt, int E, int N,
                                  const float* __restrict__ m,
                                  float* __restrict__ logit,
                                  float* __restrict__ sbuf) {
  size_t idx = (size_t)blockIdx.x * blockDim.x + threadIdx.x;
  size_t total = (size_t)(E + N) * HEADS;
  if (idx >= total) return;
  int ef = (int)(idx / HEADS);
  int h = (int)(idx % HEADS);
  int t = (ef < E) ? tgt[ef] : (ef - E);
  float mm = m[(size_t)t * HEADS + h];
  if (mm < -3.0e38f) mm = 0.0f; // where(isfinite(m), m, 0)
  float ex = expf(logit[idx] - mm);
  logit[idx] = ex;
  atomicAdd(&sbuf[(size_t)t * HEADS + h], ex);
}

// ------------------------------ GAT pass 3: alpha-weighted scatter (atomic)
template <int HEADS>
__global__ void gat_scatter_kernel(const int* __restrict__ src, const int* __restrict__ tgt,
                                   int E, int N,
                                   const float* __restrict__ logit,
                                   const float* __restrict__ sbuf,
                                   const float* __restrict__ xl,
                                   float* __restrict__ acc) {
  constexpr int HC = HEADS * 64;
  constexpr int TPL = HC / 32;
  int lane = threadIdx.x & 31;
  int wave = blockIdx.x * (blockDim.x >> 5) + (threadIdx.x >> 5);
  int nw = gridDim.x * (blockDim.x >> 5);
  int Ef = E + N;
  for (int ef = wave; ef < Ef; ef += nw) {
    int sN, tN;
    if (ef < E) { sN = src[ef]; tN = tgt[ef]; }
    else { sN = tN = ef - E; }
    float alpha[HEADS];
    #pragma unroll
    for (int h = 0; h < HEADS; h++)
      alpha[h] = logit[(size_t)ef * HEADS + h] /
                 (sbuf[(size_t)tN * HEADS + h] + 1e-16f);
    #pragma unroll
    for (int tt = 0; tt < TPL; tt++) {
      int j = lane + 32 * tt;
      atomicAdd(&acc[(size_t)tN * HC + j],
                alpha[tt >> 1] * xl[(size_t)sN * HC + j]);
    }
  }
}

// ------------------------------------------------- bias + optional ELU
__global__ void gat_finalize_kernel(const float* __restrict__ acc,
                                    const float* __restrict__ bias,
                                    float* __restrict__ dst, int N, int HC,
                                    int do_elu) {
  size_t idx = (size_t)blockIdx.x * blockDim.x + threadIdx.x;
  if (idx >= (size_t)N * HC) return;
  float v = acc[idx] + bias[idx % HC];
  if (do_elu) v = v > 0.0f ? v : (expf(v) - 1.0f);
  dst[idx] = v;
}

// ----------------------------------------------------- final output MLP
__global__ void out_mlp_kernel(const float* __restrict__ h,
                               const float* __restrict__ x,
                               const float* __restrict__ W1, const float* __restrict__ b1,
                               const float* __restrict__ W2, const float* __restrict__ b2,
                               float* __restrict__ out, int N) {
  int i = blockIdx.x * blockDim.x + threadIdx.x;
  if (i >= N) return;
  const float* hr = h + (size_t)i * 64;
  float o = b2[0];
  for (int j = 0; j < 32; j++) {
    float sum = b1[j];
    #pragma unroll 8
    for (int k = 0; k < 64; k++) sum += hr[k] * W1[k * 32 + j];
    o += fmaxf(sum, 0.0f) * W2[j];
  }
  out[i] = o * x[(size_t)i * 5];
}

// =========================================================== host driver
struct GatWeights {
  const float *Wl, *bl, *Wr, *br, *We, *att, *bias;
};

template <int DIN, int HEADS>
static void run_gat_layer(hipStream_t stream, const float* xin, int N, int E,
                          const int* src, const int* tgt, const float* eemb,
                          const float* loopat, GatWeights w, float* xl,
                          float* xr, float* logit, float* m, float* s,
                          float* acc, float* dst, bool elu) {
  constexpr int HC = HEADS * 64;
  int tiles = (N + 15) / 16;
  int gemmBlocks = (tiles + 7) / 8;
  gemm_bias_kernel<DIN, HC><<<gemmBlocks, 256, 0, stream>>>(xin, w.Wl, w.bl, xl, N);
  gemm_bias_kernel<DIN, HC><<<gemmBlocks, 256, 0, stream>>>(xin, w.Wr, w.br, xr, N);

  size_t mn = (size_t)N * HEADS;
  fill_kernel<<<(unsigned)((mn + 255) / 256), 256, 0, stream>>>(m, mn, -__builtin_inff());
  fill_kernel<<<(unsigned)((mn + 255) / 256), 256, 0, stream>>>(s, mn, 0.0f);
  size_t an = (size_t)N * HC;
  fill_kernel<<<(unsigned)((an + 255) / 256), 256, 0, stream>>>(acc, an, 0.0f);

  gat_logit_kernel<HEADS><<<2048, 256, 0, stream>>>(src, tgt, E, N, eemb, loopat,
                                                    xl, xr, w.We, w.att, logit, m);
  size_t en = (size_t)(E + N) * HEADS;
  gat_expsum_kernel<HEADS><<<(unsigned)((en + 255) / 256), 256, 0, stream>>>(
      tgt, E, N, m, logit, s);
  gat_scatter_kernel<HEADS><<<2048, 256, 0, stream>>>(src, tgt, E, N, logit, s, xl, acc);
  gat_finalize_kernel<<<(unsigned)((an + 255) / 256), 256, 0, stream>>>(
      acc, w.bias, dst, N, HC, elu ? 1 : 0);
}

extern "C" void kernel_launch(void* const* d_in, const int* in_sizes, int n_in,
                              void* d_out, int out_size, void* d_ws,
                              size_t ws_size, hipStream_t stream) {
  const float* x = (const float*)d_in[0];
  const int* eidx = (const int*)d_in[1];
  const float* edge_attr = (const float*)d_in[2];
  int N = in_sizes[0] / 5;
  int E = in_sizes[2];
  const int* src = eidx;
  const int* tgt = eidx + E;

  // parameter leaves (dict insertion order)
  const float* P[40];
  for (int i = 3; i < 40 && i < n_in; i++) P[i] = (const float*)d_in[i];

  // workspace carve-out (all f32)
  size_t off = 0;
  auto alloc = [&](size_t elems) {
    float* p = (float*)((char*)d_ws + off);
    off += elems * sizeof(float);
    return p;
  };
  float* h0 = alloc((size_t)N * 64);
  float* eemb = alloc((size_t)E * 64);
  float* loopat = alloc((size_t)N * 64);
  float* cnt = alloc((size_t)N);
  float* xl = alloc((size_t)N * 192);
  float* xr = alloc((size_t)N * 192);
  float* logit = alloc((size_t)(E + N) * 3);
  float* mbuf = alloc((size_t)N * 3);
  float* sbuf = alloc((size_t)N * 3);
  float* acc = alloc((size_t)N * 192);
  float* hbuf = alloc((size_t)N * 192);

  // 1) node feature prep + comm/toll MLP mix -> h0
  node_mlp_kernel<<<(N + 15) / 16, 32, 0, stream>>>(
      x, P[3], P[4], P[5], P[6], P[7], P[8], P[9], P[10], h0, N);

  // 2) edge MLP -> eemb
  edge_mlp_kernel<<<(E + 15) / 16, 32, 0, stream>>>(
      edge_attr, P[11], P[12], P[13], P[14], eemb, E);

  // 3) self-loop attr = segment_mean(eemb, tgt)
  fill_kernel<<<(unsigned)(((size_t)N + 255) / 256), 256, 0, stream>>>(cnt, N, 0.0f);
  fill_kernel<<<(unsigned)(((size_t)N * 64 + 255) / 256), 256, 0, stream>>>(
      loopat, (size_t)N * 64, 0.0f);
  loop_scatter_kernel<<<(unsigned)(((size_t)E * 64 + 255) / 256), 256, 0, stream>>>(
      tgt, eemb, loopat, cnt, E);
  loop_div_kernel<<<(unsigned)(((size_t)N * 64 + 255) / 256), 256, 0, stream>>>(
      loopat, cnt, N);

  // 4) conv1 (din=64, heads=3) + ELU -> hbuf
  GatWeights w1 = {P[15], P[16], P[17], P[18], P[19], P[20], P[21]};
  run_gat_layer<64, 3>(stream, h0, N, E, src, tgt, eemb, loopat, w1, xl, xr,
                       logit, mbuf, sbuf, acc, hbuf, true);

  // 5) conv2 (din=192, heads=3) + ELU -> hbuf (in-place is safe: GEMMs read first)
  GatWeights w2 = {P[22], P[23], P[24], P[25], P[26], P[27], P[28]};
  run_gat_layer<192, 3>(stream, hbuf, N, E, src, tgt, eemb, loopat, w2, xl, xr,
                        logit, mbuf, sbuf, acc, hbuf, true);

  // 6) conv3 (din=192, heads=1, no ELU) -> h0
  GatWeights w3 = {P[29], P[30], P[31], P[32], P[33], P[34], P[35]};
  run_gat_layer<192, 1>(stream, hbuf, N, E, src, tgt, eemb, loopat, w3, xl, xr,
                        logit, mbuf, sbuf, acc, h0, false);

  // 7) output MLP * mask -> d_out
  out_mlp_kernel<<<(N + 255) / 256, 256, 0, stream>>>(
      h0, x, P[36], P[37], P[38], P[39], (float*)d_out, N);
}